// RGBDVideoTower_50251117363369
// MI455X (gfx1250) — compile-verified
//
#include <hip/hip_runtime.h>
#include <hip/hip_bf16.h>
#include <limits.h>

// Problem constants (match the reference)
#define BATCH 8
#define NPTS  9216
#define FEAT  1024
#define KVOX  4096
#define T1    1024
#define NPT   (NPTS / T1)      // 9 points per thread in build kernel
#define T2    256              // pool kernel: 256 threads * float4 = 1024 cols

// ---------------------------------------------------------------------------
// Kernel 1: per-batch voxelization. One 1024-thread workgroup per batch.
//   - per-axis min of floor(xyz/0.2)
//   - compact voxel id cid = (cx<<8)|(cy<<4)|cz  (rank-equivalent to the
//     reference's (cx*1024+cy)*1024+cz hash since coords < 16)
//   - LDS histogram over 4096 bins, dual block prefix-scan:
//       occupancy scan -> compacted voxel rank
//       count scan     -> offsets into per-batch point-index list
//   - scatter point indices into plist, write vstart/vcnt per rank slot
// ---------------------------------------------------------------------------
__global__ __launch_bounds__(T1) void voxel_build_kernel(
    const float* __restrict__ xyz, int* __restrict__ plist,
    int* __restrict__ vstart, int* __restrict__ vcnt,
    int* __restrict__ counts)
{
  __shared__ int s_hist[KVOX];
  __shared__ int s_cursor[KVOX];
  __shared__ int s_scanA[T1];
  __shared__ int s_scanB[T1];
  __shared__ int s_min3[3];

  const int b   = blockIdx.x;
  const int tid = threadIdx.x;
  const float* xp = xyz + (size_t)b * NPTS * 3;

  int cid[NPT];
  int cxl[NPT], cyl[NPT], czl[NPT];
  int mnx = INT_MAX, mny = INT_MAX, mnz = INT_MAX;
  for (int k = 0; k < NPT; ++k) {
    const int i = tid + k * T1;
    const float fx = xp[(size_t)i * 3 + 0];
    const float fy = xp[(size_t)i * 3 + 1];
    const float fz = xp[(size_t)i * 3 + 2];
    const int cx = (int)floorf(fx / 0.2f);
    const int cy = (int)floorf(fy / 0.2f);
    const int cz = (int)floorf(fz / 0.2f);
    cxl[k] = cx; cyl[k] = cy; czl[k] = cz;
    mnx = min(mnx, cx); mny = min(mny, cy); mnz = min(mnz, cz);
  }
  for (int c = tid; c < KVOX; c += T1) s_hist[c] = 0;
  if (tid < 3) s_min3[tid] = INT_MAX;
  __syncthreads();
  atomicMin(&s_min3[0], mnx);
  atomicMin(&s_min3[1], mny);
  atomicMin(&s_min3[2], mnz);
  __syncthreads();
  mnx = s_min3[0]; mny = s_min3[1]; mnz = s_min3[2];

  for (int k = 0; k < NPT; ++k) {
    const int ax = min(cxl[k] - mnx, 15);
    const int ay = min(cyl[k] - mny, 15);
    const int az = min(czl[k] - mnz, 15);
    cid[k] = (ax << 8) | (ay << 4) | az;
    atomicAdd(&s_hist[cid[k]], 1);
  }
  __syncthreads();

  // Each thread owns 4 consecutive bins; local occupancy & count sums.
  int occ = 0, csum = 0;
  const int bin0 = tid * (KVOX / T1);
  for (int j = 0; j < KVOX / T1; ++j) {
    const int h = s_hist[bin0 + j];
    occ  += (h > 0);
    csum += h;
  }
  s_scanA[tid] = occ; s_scanB[tid] = csum;
  __syncthreads();
  // Hillis-Steele inclusive block scan over both arrays.
  for (int off = 1; off < T1; off <<= 1) {
    int a = 0, c = 0;
    if (tid >= off) { a = s_scanA[tid - off]; c = s_scanB[tid - off]; }
    __syncthreads();
    if (tid >= off) { s_scanA[tid] += a; s_scanB[tid] += c; }
    __syncthreads();
  }
  int occ_base = s_scanA[tid] - occ;     // exclusive occupancy prefix
  int cnt_base = s_scanB[tid] - csum;    // exclusive count prefix
  const int total_occ = s_scanA[T1 - 1];

  // Zero vcnt for all K slots (empty slots must read as 0 in kernel 2).
  for (int r = tid; r < KVOX; r += T1) vcnt[(size_t)b * KVOX + r] = 0;
  __syncthreads();

  for (int j = 0; j < KVOX / T1; ++j) {
    const int c = bin0 + j;
    const int h = s_hist[c];
    s_cursor[c] = cnt_base;                      // point-list write cursor
    if (h > 0) {
      vstart[(size_t)b * KVOX + occ_base] = cnt_base;
      vcnt  [(size_t)b * KVOX + occ_base] = h;
      ++occ_base;
    }
    cnt_base += h;
  }
  if (tid == 0) counts[b] = total_occ;
  __syncthreads();

  // Scatter point indices grouped by voxel (within-voxel order is made
  // deterministic later by kernel 2's ascending-index selection).
  for (int k = 0; k < NPT; ++k) {
    const int pos = atomicAdd(&s_cursor[cid[k]], 1);
    plist[(size_t)b * NPTS + pos] = tid + k * T1;
  }
}

// ---------------------------------------------------------------------------
// Kernel 2: one workgroup per output voxel row. Double-buffered gather of
// 4 KB feature rows via CDNA5 async global->LDS DMA (ASYNCcnt), then
// register accumulation and mean. Empty rows write zeros.
// ---------------------------------------------------------------------------
__global__ __launch_bounds__(T2) void voxel_pool_kernel(
    const float* __restrict__ feats, const int* __restrict__ plist,
    const int* __restrict__ vstart, const int* __restrict__ vcnt,
    float* __restrict__ out)
{
  __shared__ float smem[2 * FEAT];   // two 4 KB staging buffers

  const int row = blockIdx.x;        // 0 .. B*K-1
  const int tid = threadIdx.x;
  const int cnt = vcnt[row];
  float4* orow = reinterpret_cast<float4*>(out + (size_t)row * FEAT) + tid;
  if (cnt <= 0) { *orow = make_float4(0.f, 0.f, 0.f, 0.f); return; }

  const int b     = row >> 12;       // row / KVOX
  const int start = vstart[row];
  const int* pl   = plist + (size_t)b * NPTS;

  // Low 32 bits of a generic pointer to __shared__ == wave-relative LDS
  // byte offset (flat aperture mapping).
  const unsigned lds0 = (unsigned)(unsigned long long)(uintptr_t)(smem + tid * 4);
  const unsigned lds1 = lds0 + (unsigned)(FEAT * sizeof(float));
  const float* gbase  = feats + (size_t)b * NPTS * FEAT + tid * 4;

  // Deterministic ascending point-index order within the voxel (selection
  // over the unordered list; cnt averages ~2-3 here, so O(cnt^2) is free).
  auto sel_min_gt = [&](int prev) -> int {
    int cur = INT_MAX;
    for (int t = 0; t < cnt; ++t) {
      const int v = pl[start + t];
      if (v > prev && v < cur) cur = v;
    }
    return cur;
  };

  int cur = sel_min_gt(-1);
  asm volatile("global_load_async_to_lds_b128 %0, %1, off"
               :: "v"(lds0), "v"(gbase + (size_t)cur * FEAT) : "memory");

  float4 acc = make_float4(0.f, 0.f, 0.f, 0.f);
  for (int j = 0; j < cnt; ++j) {
    const float* sread = (j & 1) ? (smem + FEAT) : smem;
    if (j + 1 < cnt) {
      const int nxt = sel_min_gt(cur);
      const unsigned ldsn = (j & 1) ? lds0 : lds1;
      asm volatile("global_load_async_to_lds_b128 %0, %1, off"
                   :: "v"(ldsn), "v"(gbase + (size_t)nxt * FEAT) : "memory");
      asm volatile("s_wait_asynccnt 1" ::: "memory");  // oldest DMA done
      cur = nxt;
    } else {
      asm volatile("s_wait_asynccnt 0" ::: "memory");  // drain
    }
    const float4 v = *(reinterpret_cast<const float4*>(sread) + tid);
    acc.x += v.x; acc.y += v.y; acc.z += v.z; acc.w += v.w;
  }
  const float c = (float)cnt;        // cnt >= 1 here; matches max(cnt,1)
  acc.x /= c; acc.y /= c; acc.z /= c; acc.w /= c;
  *orow = acc;
}

// ---------------------------------------------------------------------------
// Kernel 3: batch_offset = cumsum(counts), written as float at tail of d_out.
// ---------------------------------------------------------------------------
__global__ void batch_offset_kernel(const int* __restrict__ counts,
                                    float* __restrict__ out_off)
{
  if (blockIdx.x == 0 && threadIdx.x == 0) {
    int s = 0;
    for (int b = 0; b < BATCH; ++b) {
      s += counts[b];
      out_off[b] = (float)s;
    }
  }
}

extern "C" void kernel_launch(void* const* d_in, const int* in_sizes, int n_in,
                              void* d_out, int out_size, void* d_ws, size_t ws_size,
                              hipStream_t stream) {
  (void)in_sizes; (void)n_in; (void)out_size; (void)ws_size;
  const float* feats = (const float*)d_in[0];   // (B, N, F) float32
  const float* xyz   = (const float*)d_in[1];   // (B, N, 3) float32
  float* out = (float*)d_out;                   // (B*K*F) pooled + (B) offsets

  // Workspace layout (~557 KB of d_ws):
  int* plist  = (int*)d_ws;                     // B*N point indices, voxel-grouped
  int* vstart = plist  + BATCH * NPTS;          // B*K list offsets
  int* vcnt   = vstart + BATCH * KVOX;          // B*K per-slot counts
  int* counts = vcnt   + BATCH * KVOX;          // B occupied-voxel counts

  voxel_build_kernel<<<BATCH, T1, 0, stream>>>(xyz, plist, vstart, vcnt, counts);
  voxel_pool_kernel<<<BATCH * KVOX, T2, 0, stream>>>(feats, plist, vstart, vcnt, out);
  batch_offset_kernel<<<1, 32, 0, stream>>>(counts, out + (size_t)BATCH * KVOX * FEAT);
}